// MonarchTransition_62302795596622
// MI455X (gfx1250) — compile-verified
//
#include <hip/hip_runtime.h>
#include <math.h>

// MonarchTransition on MI455X (gfx1250), wave32, f32 WMMA 16x16x4.
//
// batch=8192, d=1024, b=64, n=4096.
// ws layout (floats): QtPR[64*64*64] | QtPL[64*64*64] | gates[8192*128] (=6MB)
//
// All B-side operands are stored "K-pair interleaved":
//   Buf[kp][2*n + e] == B[k = 2*kp + e][n]
// so a WMMA 16x16x4 B-fragment (lane: k = k0+2*half, k0+2*half+1; n = n0+lr)
// is ONE aligned 8-byte load into a contiguous VGPR pair (no fixup moves).

typedef __attribute__((ext_vector_type(2))) float v2f;
typedef __attribute__((ext_vector_type(4))) float v4f;
typedef __attribute__((ext_vector_type(8))) float v8f;

#define QTR_OFF   0
#define QTL_OFF   (64 * 64 * 64)
#define GATES_OFF (2 * 64 * 64 * 64)

static __device__ __forceinline__ v8f v8f_zero() {
  v8f z;
#pragma unroll
  for (int e = 0; e < 8; ++e) z[e] = 0.0f;
  return z;
}

// D = A(16x4 f32) * B(4x16 f32) + C(16x16 f32)
static __device__ __forceinline__ v8f wmma4(v2f a, v2f b, v8f c) {
  return __builtin_amdgcn_wmma_f32_16x16x4_f32(false, a, false, b, (short)0, c,
                                               false, false);
}

// ---------------------------------------------------------------------------
// Kernel 1: Cayley transform. Q = (I+A)^-1 (I-A), A = S - S^T, per c-slice.
// Gauss-Jordan in LDS (I+A = I + skew => always nonsingular, no pivoting).
// Stores Q^T in K-pair-interleaved layout: QtP[kp*128 + 2*b + e] = Q[b][2kp+e].
// blockIdx.x = c (0..63), blockIdx.y = 0 -> R, 1 -> L. 256 threads.
// ---------------------------------------------------------------------------
__global__ __launch_bounds__(256) void cayley_kernel(
    const float* __restrict__ Lskew, const float* __restrict__ Rskew,
    float* __restrict__ ws) {
  __shared__ float Ms[64][66];
  __shared__ float Ns[64][66];
  const int c = blockIdx.x;
  const float* S = (blockIdx.y == 0 ? Rskew : Lskew) + c * 4096;
  float* Qt = ws + (blockIdx.y == 0 ? QTR_OFF : QTL_OFF) + c * 4096;
  const int t = threadIdx.x;

  // M = I + A, N = I - A
  for (int idx = t; idx < 4096; idx += 256) {
    int i = idx >> 6, j = idx & 63;
    float a = S[i * 64 + j] - S[j * 64 + i];
    float e = (i == j) ? 1.0f : 0.0f;
    Ms[i][j] = e + a;
    Ns[i][j] = e - a;
  }
  __syncthreads();

  const int i = t >> 2;           // row this thread eliminates
  const int jb = (t & 3) << 4;    // 16-column slice
  for (int k = 0; k < 64; ++k) {
    float pinv = 1.0f / Ms[k][k];   // uniform read
    __syncthreads();
    if (t < 64) { Ms[k][t] *= pinv; Ns[k][t] *= pinv; }
    __syncthreads();
    float f = Ms[i][k];             // factor, read before row updates
    __syncthreads();
    if (i != k) {
#pragma unroll
      for (int jj = 0; jj < 16; ++jj) {
        int j = jb + jj;
        Ms[i][j] -= f * Ms[k][j];
        Ns[i][j] -= f * Ns[k][j];
      }
    }
    __syncthreads();
  }

  // N holds Q ([b][j]); store pair-interleaved Q^T:
  // QtP[kp*128 + 2*b + e] = Q[b][2*kp + e]
  for (int idx = t; idx < 4096; idx += 256) {
    int e = idx & 1, b = (idx >> 1) & 63, kp = idx >> 7;
    Qt[idx] = Ns[b][kp * 2 + e];
  }
}

// ---------------------------------------------------------------------------
// Kernel 2: gates = sigmoid(x @ gate_W^T + gate_b), (8192x1024)@(1024x128).
// One block = 64 batch rows x 128 cols, 128 threads (4 waves).
// ---------------------------------------------------------------------------
__global__ __launch_bounds__(128) void gates_kernel(
    const float* __restrict__ x, const float* __restrict__ gw,
    const float* __restrict__ gb, float* __restrict__ gates) {
  __shared__ float Xs[64][68];    // [batch][k], pad 68: conflict-free A b64
  __shared__ float Ws[32][288];   // [kp][2n+e], stride 288 = 32 mod 64
  const int t = threadIdx.x;
  const int bb = blockIdx.x * 64;
  const int lane = t & 31, wv = t >> 5;
  const int half = lane >> 4, lr = lane & 15;
  const int m0 = wv * 16;

  v8f acc[8];
#pragma unroll
  for (int nt = 0; nt < 8; ++nt) acc[nt] = v8f_zero();

  for (int kb = 0; kb < 16; ++kb) {
    for (int idx = t; idx < 1024; idx += 128) {     // stage X (coalesced)
      int r = idx >> 4, q = (idx & 15) * 4;
      *(v4f*)&Xs[r][q] = *(const v4f*)(x + (size_t)(bb + r) * 1024 + kb * 64 + q);
    }
    {                                               // stage W^T pair-interleaved
      int n = t;
#pragma unroll
      for (int kq = 0; kq < 16; ++kq) {             // k = 4*kq .. 4*kq+3
        v4f v = *(const v4f*)(gw + (size_t)n * 1024 + kb * 64 + kq * 4);
        v2f lo; lo.x = v.x; lo.y = v.y;             // kp = 2*kq
        v2f hi; hi.x = v.z; hi.y = v.w;             // kp = 2*kq+1
        *(v2f*)&Ws[kq * 2 + 0][2 * n] = lo;
        *(v2f*)&Ws[kq * 2 + 1][2 * n] = hi;
      }
    }
    __syncthreads();
#pragma unroll
    for (int k0 = 0; k0 < 64; k0 += 4) {
      int kp = (k0 >> 1) + half;
      v2f a = *(const v2f*)&Xs[m0 + lr][k0 + 2 * half];
#pragma unroll
      for (int nt = 0; nt < 8; ++nt) {
        v2f b = *(const v2f*)&Ws[kp][2 * (nt * 16 + lr)];
        acc[nt] = wmma4(a, b, acc[nt]);
      }
    }
    __syncthreads();
  }

#pragma unroll
  for (int nt = 0; nt < 8; ++nt) {
    int n = nt * 16 + lr;
    float bias = gb[n];
#pragma unroll
    for (int r = 0; r < 8; ++r) {
      int m = m0 + r + 8 * half;
      float val = acc[nt][r] + bias;
      gates[(size_t)(bb + m) * 128 + n] = 1.0f / (1.0f + __expf(-val));
    }
  }
}

// ---------------------------------------------------------------------------
// Kernel 3 (R-pass): out[B, c*64+b] = beta[B,c] * sum_j h[B, c*64+j]*Q_R[c][b][j]
// Block = (c, 64-batch tile), 128 threads. Writes intermediate INTO d_out.
// ---------------------------------------------------------------------------
__global__ __launch_bounds__(128) void rpass_kernel(
    const float* __restrict__ h, const float* __restrict__ ws,
    float* __restrict__ out) {
  __shared__ float Qs[32][160];   // QtP_R[c]: [kp][2b+e], stride 160 = 32 mod 64
  __shared__ float Hs[64][68];    // [batch][j], padded
  const int c = blockIdx.x;
  const int bb = blockIdx.y * 64;
  const float* qt = ws + QTR_OFF + c * 4096;
  const float* gates = ws + GATES_OFF;
  const int t = threadIdx.x;
  const int lane = t & 31, wv = t >> 5;
  const int half = lane >> 4, lr = lane & 15;
  const int m0 = wv * 16;

  for (int idx = t; idx < 1024; idx += 128) {       // flat copy of QtP rows
    int row = idx >> 5, q = (idx & 31) * 4;
    *(v4f*)&Qs[row][q] = *(const v4f*)(qt + row * 128 + q);
  }
  for (int idx = t; idx < 1024; idx += 128) {
    int r = idx >> 4, q = (idx & 15) * 4;
    *(v4f*)&Hs[r][q] = *(const v4f*)(h + (size_t)(bb + r) * 4096 + c * 64 + q);
  }
  __syncthreads();

  v8f acc[4];
#pragma unroll
  for (int nt = 0; nt < 4; ++nt) acc[nt] = v8f_zero();
#pragma unroll
  for (int k0 = 0; k0 < 64; k0 += 4) {
    int kp = (k0 >> 1) + half;
    v2f a = *(const v2f*)&Hs[m0 + lr][k0 + 2 * half];
#pragma unroll
    for (int nt = 0; nt < 4; ++nt) {
      v2f b = *(const v2f*)&Qs[kp][2 * (nt * 16 + lr)];
      acc[nt] = wmma4(a, b, acc[nt]);
    }
  }

#pragma unroll
  for (int nt = 0; nt < 4; ++nt) {
#pragma unroll
    for (int r = 0; r < 8; ++r) {
      int m = m0 + r + 8 * half;
      float beta = gates[(size_t)(bb + m) * 128 + 64 + c];
      out[(size_t)(bb + m) * 4096 + c * 64 + nt * 16 + lr] = beta * acc[nt][r];
    }
  }
}

// ---------------------------------------------------------------------------
// Kernel 4 (L-pass, IN PLACE on d_out): per column c' of each 64x64 block:
//   w[b] = alpha[B,c'] * sum_j Q_L[c'][b][j] * v[j],  v[j] = out[B, j*64+c']
// Output addresses == input addresses per (B,c') => safe in place per block.
// Block = (16 consecutive c', 16-batch tile). 64KB LDS absorbs the transpose.
// Vs[cc][jp][B][e] holds v[2*jp+e]: A-fragment = one aligned 8B LDS load,
// jp-stride = 32 words => half-waves hit disjoint bank halves.
// ---------------------------------------------------------------------------
__global__ __launch_bounds__(128) void lpass_kernel(
    float* __restrict__ out, const float* __restrict__ ws) {
  __shared__ float Vs[16][32][16][2];   // [cc][jp][B][e], 64KB
  const int c0 = blockIdx.x * 16;
  const int bb = blockIdx.y * 16;
  const float* qtl = ws + QTL_OFF;
  const float* gates = ws + GATES_OFF;
  const int t = threadIdx.x;
  const int lane = t & 31, wv = t >> 5;
  const int half = lane >> 4, lr = lane & 15;

  // Stage columns c0..c0+15: 64B-contiguous float4 global loads over cc.
  for (int idx = t; idx < 4096; idx += 128) {
    int ccq = idx & 3, j = (idx >> 2) & 63, B = idx >> 8;
    v4f v = *(const v4f*)(out + (size_t)(bb + B) * 4096 + j * 64 + c0 + ccq * 4);
    int jp = j >> 1, jo = j & 1;
    Vs[ccq * 4 + 0][jp][B][jo] = v.x; Vs[ccq * 4 + 1][jp][B][jo] = v.y;
    Vs[ccq * 4 + 2][jp][B][jo] = v.z; Vs[ccq * 4 + 3][jp][B][jo] = v.w;
  }
  __syncthreads();

  // Each wave owns 4 columns cc = wv*4 .. wv*4+3
#pragma unroll
  for (int q = 0; q < 4; ++q) {
    const int cc = wv * 4 + q;
    const int cp = c0 + cc;
    const float* Q = qtl + (size_t)cp * 4096;   // QtP_L[cp]: [kp][2b+e]
    v8f acc[4];
#pragma unroll
    for (int nt = 0; nt < 4; ++nt) acc[nt] = v8f_zero();
#pragma unroll
    for (int k0 = 0; k0 < 64; k0 += 4) {
      int kp = (k0 >> 1) + half;
      v2f a = *(const v2f*)&Vs[cc][kp][lr][0];            // one ds_load_b64
#pragma unroll
      for (int nt = 0; nt < 4; ++nt) {
        v2f b = *(const v2f*)(Q + kp * 128 + 2 * (nt * 16 + lr));  // b64 global
        acc[nt] = wmma4(a, b, acc[nt]);
      }
    }
    // alpha scale, write back into the (now fully consumed) cc region,
    // reinterpreting jp/e as output-row pair index: Vs[cc][b>>1][B][b&1]
#pragma unroll
    for (int nt = 0; nt < 4; ++nt) {
#pragma unroll
      for (int r = 0; r < 8; ++r) {
        int m = r + 8 * half;
        int n = nt * 16 + lr;
        float alpha = gates[(size_t)(bb + m) * 128 + cp];
        Vs[cc][n >> 1][m][n & 1] = alpha * acc[nt][r];
      }
    }
  }
  __syncthreads();

  // Coalesced float4 stores back to the same addresses.
  for (int idx = t; idx < 4096; idx += 128) {
    int ccq = idx & 3, b = (idx >> 2) & 63, B = idx >> 8;
    int bp = b >> 1, bo = b & 1;
    v4f v;
    v.x = Vs[ccq * 4 + 0][bp][B][bo]; v.y = Vs[ccq * 4 + 1][bp][B][bo];
    v.z = Vs[ccq * 4 + 2][bp][B][bo]; v.w = Vs[ccq * 4 + 3][bp][B][bo];
    *(v4f*)(out + (size_t)(bb + B) * 4096 + b * 64 + c0 + ccq * 4) = v;
  }
}

// ---------------------------------------------------------------------------
extern "C" void kernel_launch(void* const* d_in, const int* in_sizes, int n_in,
                              void* d_out, int out_size, void* d_ws,
                              size_t ws_size, hipStream_t stream) {
  const float* x    = (const float*)d_in[0];   // (8192, 1024)
  const float* h    = (const float*)d_in[1];   // (8192, 4096)
  const float* Lsk  = (const float*)d_in[2];   // (64, 64, 64)
  const float* Rsk  = (const float*)d_in[3];   // (64, 64, 64)
  const float* gw   = (const float*)d_in[4];   // (128, 1024)
  const float* gb   = (const float*)d_in[5];   // (128,)
  float* out = (float*)d_out;                  // (8192, 4096)
  float* ws  = (float*)d_ws;                   // needs >= 6 MB

  // 1) Cayley transforms -> pair-interleaved Qt_R, Qt_L in ws
  cayley_kernel<<<dim3(64, 2), 256, 0, stream>>>(Lsk, Rsk, ws);
  // 2) gates = sigmoid(x @ W^T + b) -> ws
  gates_kernel<<<dim3(128), 128, 0, stream>>>(x, gw, gb, ws + GATES_OFF);
  // 3) R-pass (beta applied), intermediate written into d_out
  rpass_kernel<<<dim3(64, 128), 128, 0, stream>>>(h, ws, out);
  // 4) L-pass (alpha applied), in place on d_out (column-local transform)
  lpass_kernel<<<dim3(4, 512), 128, 0, stream>>>(out, ws);
}